// attention1_18124761989893
// MI455X (gfx1250) — compile-verified
//
#include <hip/hip_runtime.h>

#define N_NODES 128
#define NR_EDGES (N_NODES * (N_NODES - 1))   // 16256 (divisible by 4)
#define FPRIME 32
#define F_DIM 16
#define B_DIM 64

typedef __attribute__((ext_vector_type(2))) float v2f;
typedef __attribute__((ext_vector_type(8))) float v8f;

// Kernel 1: fold attention vector into projection: wa[sel][f] = sum_g a[sel*32+g] * w[g,f]
__global__ void __launch_bounds__(32) fuse_aw_kernel(const float* __restrict__ w,
                                                     const float* __restrict__ a,
                                                     float* __restrict__ wa) {
    const int t   = threadIdx.x;
    const int f   = t & 15;
    const int sel = t >> 4;
    float acc = 0.f;
#pragma unroll
    for (int g = 0; g < FPRIME; ++g)
        acc = fmaf(a[sel * FPRIME + g], w[g * F_DIM + f], acc);
    wa[sel * F_DIM + f] = acc;
}

// Kernel 2: per (b, 16-node tile) compute p[b,n] = x[b,:,n]·wa1, q[b,n] = x[b,:,n]·wa2
// via V_WMMA_F32_16X16X4_F32 (K=16 folded as 4 slabs of K=4).
//   A (16x4 f32): lanes 0-15 hold M=0..15 with {K=0,K=1}; lanes 16-31 hold {K=2,K=3}
//   B (4x16 f32): lanes 0-15 hold N with rows {K=0,K=1}; lanes 16-31 rows {K=2,K=3}
//   D (16x16 f32): VGPR r, lanes 0-15 -> M=r, N=lane; lanes 16-31 -> M=r+8
__global__ void __launch_bounds__(32) node_pq_kernel(const float* __restrict__ x,
                                                     const float* __restrict__ wa,
                                                     float* __restrict__ p,
                                                     float* __restrict__ q) {
    const int b     = blockIdx.x >> 3;          // batch
    const int n0    = (blockIdx.x & 7) << 4;    // node tile base
    const int l     = threadIdx.x;
    const int mA    = l & 15;                   // A-matrix row (node within tile)
    const int hi    = l >> 4;                   // lane half
    const int kHalf = hi << 1;                  // K offset within K=4 slab (0 or 2)
    const int col   = l & 15;                   // B/D column
    const bool live = (col < 2);
    const int  colc = live ? col : 0;           // clamped: ALL lanes load valid addrs

    const float* xb  = x + (size_t)b * F_DIM * N_NODES;
    const float* war = wa + colc * F_DIM;       // this lane's wa row (wa1 or wa2)

    // Hoist all B-fragment loads: unconditional aligned b64 pairs, one clause,
    // then value-level select (v_cndmask) — no EXEC-predicated loads.
    v2f bp[4];
#pragma unroll
    for (int s = 0; s < 4; ++s)
        bp[s] = *(const v2f*)(war + 4 * s + kHalf);   // 8B-aligned pair {f, f+1}

    v8f c = {};
#pragma unroll
    for (int s = 0; s < 4; ++s) {
        const int kA = 4 * s + kHalf;           // f index of this lane's K pair
        v2f av, bv;
        av.x = xb[(kA + 0) * N_NODES + n0 + mA];
        av.y = xb[(kA + 1) * N_NODES + n0 + mA];
        bv.x = live ? bp[s].x : 0.f;
        bv.y = live ? bp[s].y : 0.f;
        // 8 args: (neg_a, A, neg_b, B, c_mod, C, reuse_a, reuse_b)
        c = __builtin_amdgcn_wmma_f32_16x16x4_f32(false, av, false, bv,
                                                  (short)0, c, false, false);
    }

    // D column 0 = p, column 1 = q; rows M = mBase + r live in c[r]
    const int mBase = hi << 3;
    if (live) {
        float* dst = ((col == 0) ? p : q) + (size_t)b * N_NODES + n0 + mBase;
#pragma unroll
        for (int r = 0; r < 4; ++r) {
            v2f o2 = { c[2 * r], c[2 * r + 1] };
            *(v2f*)(dst + 2 * r) = o2;          // b64 stores of contiguous D rows
        }
    }
}

// Kernel 3: out[b,e] = p[b,recv[e]] + q[b,send[e]]  (store-bandwidth bound)
// 4 edges per thread: int4 index loads, 8 L2-resident gathers, one b128 store.
__global__ void __launch_bounds__(256) edge_gather_kernel(const float* __restrict__ p,
                                                          const float* __restrict__ q,
                                                          const int* __restrict__ recv,
                                                          const int* __restrict__ send,
                                                          float* __restrict__ out) {
    const int e4 = blockIdx.x * 256 + threadIdx.x;   // quad-edge index
    const int b  = blockIdx.y;
    if (e4 < NR_EDGES / 4) {
        const int4 r4 = ((const int4*)recv)[e4];
        const int4 s4 = ((const int4*)send)[e4];
        const float* pb = p + b * N_NODES;           // 512B table, cache-resident
        const float* qb = q + b * N_NODES;
        float4 o;
        o.x = pb[r4.x] + qb[s4.x];
        o.y = pb[r4.y] + qb[s4.y];
        o.z = pb[r4.z] + qb[s4.z];
        o.w = pb[r4.w] + qb[s4.w];
        ((float4*)(out + (size_t)b * NR_EDGES))[e4] = o;
    }
}

extern "C" void kernel_launch(void* const* d_in, const int* in_sizes, int n_in,
                              void* d_out, int out_size, void* d_ws, size_t ws_size,
                              hipStream_t stream) {
    const float* x    = (const float*)d_in[0];   // (64,16,128)
    const float* w    = (const float*)d_in[1];   // (32,16)
    const float* a    = (const float*)d_in[2];   // (1,64)
    const int*   recv = (const int*)d_in[3];     // (16256,)
    const int*   send = (const int*)d_in[4];     // (16256,)
    float* out = (float*)d_out;                  // (64,16256,1)

    // workspace: wa[32] | p[64*128] | q[64*128]  = ~64.1 KB of floats
    float* wa = (float*)d_ws;
    float* p  = wa + 32;
    float* q  = p + B_DIM * N_NODES;

    fuse_aw_kernel<<<1, 32, 0, stream>>>(w, a, wa);
    node_pq_kernel<<<B_DIM * (N_NODES / 16), 32, 0, stream>>>(x, wa, p, q);
    dim3 g3(((NR_EDGES / 4) + 255) / 256, B_DIM);
    edge_gather_kernel<<<g3, 256, 0, stream>>>(p, q, recv, send, out);
}